// PhysicalSlidingAttention_78365973283102
// MI455X (gfx1250) — compile-verified
//
#include <hip/hip_runtime.h>
#include <hip/hip_bf16.h>

typedef float v2f __attribute__((ext_vector_type(2)));
typedef float v8f __attribute__((ext_vector_type(8)));

#define D_MODELC 1024
#define N_HEADSC 16
#define D_KC     64
#define BSZ      8
#define LQ       512
#define LK       512
#define PAD_POSF 10000.0f

// ---------------------------------------------------------------------------
// Generic batched WMMA fp32 GEMM:  C = alpha * A @ op(B) + bias
//   A: [M,K] row-major (lda);  B: bT ? [N,K] : [K,N] (ldb);  C: [M,N] (ldc)
//   batch z -> (zb = z/hdiv, zh = z%hdiv); per-level pointer strides.
// Tile: 128x64 per workgroup (128 threads = 4 waves), BK=16.
// Each wave: 32 rows x 64 cols = 8 v8f accumulators via v_wmma_f32_16x16x4_f32.
// LDS layout [output-row][k] with row stride BK+4 so every fragment is one
// 8B-aligned, bank-conflict-free ds_load_b64.
// ---------------------------------------------------------------------------
#define BM 128
#define BN 64
#define BK 16
#define LDSP (BK + 4)   // row pitch in floats: keeps b64 frags aligned + conflict-free

__global__ __launch_bounds__(128)
void wmma_gemm_kernel(const float* __restrict__ A, const float* __restrict__ B,
                      const float* __restrict__ bias, float* __restrict__ C,
                      int M, int N, int K, int lda, int ldb, int ldc, int bT,
                      float alpha, int hdiv,
                      long long sAb, long long sAh, long long sBb, long long sBh,
                      long long sCb, long long sCh)
{
  __shared__ float As[BM][LDSP];   // [m][k]
  __shared__ float Bs[BN][LDSP];   // [n][k]  (B held transposed in LDS)

  const int z  = blockIdx.z;
  const long long zb = z / hdiv, zh = z % hdiv;
  A += (size_t)(zb * sAb + zh * sAh);
  B += (size_t)(zb * sBb + zh * sBh);
  C += (size_t)(zb * sCb + zh * sCh);

  const int nBase = blockIdx.x * BN;
  const int mBase = blockIdx.y * BM;
  const int tid  = threadIdx.x;
  const int lane = tid & 31;
  const int wave = tid >> 5;      // 0..3
  const int half = lane >> 4;     // 0/1
  const int l16  = lane & 15;

  const v8f z8 = {0.f,0.f,0.f,0.f,0.f,0.f,0.f,0.f};
  v8f acc[2][4];
#pragma unroll
  for (int s = 0; s < 2; ++s)
#pragma unroll
    for (int nt = 0; nt < 4; ++nt) acc[s][nt] = z8;

  for (int k0 = 0; k0 < K; k0 += BK) {
    // ---- stage A tile (128x16): 512 float4, 4 per thread, b128 LDS stores ----
#pragma unroll
    for (int i = 0; i < 4; ++i) {
      const int f  = tid + i * 128;
      const int r  = f >> 2;            // m within tile: 0..127
      const int c4 = (f & 3) << 2;      // k within tile: 0,4,8,12
      const float4 v = *reinterpret_cast<const float4*>(A + (size_t)(mBase + r) * lda + k0 + c4);
      *reinterpret_cast<float4*>(&As[r][c4]) = v;
    }
    // ---- stage B tile into Bs[n][k] ----
    if (bT) {   // B is [N,K]: direct copy, no transpose
#pragma unroll
      for (int i = 0; i < 2; ++i) {
        const int f  = tid + i * 128;
        const int r  = f >> 2;          // n within tile: 0..63
        const int c4 = (f & 3) << 2;    // k within tile
        const float4 v = *reinterpret_cast<const float4*>(B + (size_t)(nBase + r) * ldb + k0 + c4);
        *reinterpret_cast<float4*>(&Bs[r][c4]) = v;
      }
    } else {    // B is [K,N]: transpose on store
#pragma unroll
      for (int i = 0; i < 2; ++i) {
        const int f  = tid + i * 128;
        const int r  = f >> 4;          // k within tile: 0..15
        const int c4 = (f & 15) << 2;   // n within tile: 0..60
        const float4 v = *reinterpret_cast<const float4*>(B + (size_t)(k0 + r) * ldb + nBase + c4);
        Bs[c4 + 0][r] = v.x; Bs[c4 + 1][r] = v.y; Bs[c4 + 2][r] = v.z; Bs[c4 + 3][r] = v.w;
      }
    }
    __syncthreads();

    // prefetch next K-chunk while we compute (global_prefetch_b8)
    if (k0 + BK < K) {
      __builtin_prefetch(A + (size_t)(mBase + tid) * lda + (k0 + BK), 0, 0);
      if (bT)
        __builtin_prefetch(B + (size_t)(nBase + (tid >> 1)) * ldb + (k0 + BK) + (tid & 1) * 8, 0, 0);
      else
        __builtin_prefetch(B + (size_t)(k0 + BK + (tid >> 3)) * ldb + nBase + (tid & 7) * 8, 0, 0);
    }

    const int arow0 = wave * 32 + l16;        // first 16-row subtile
    const int arow1 = arow0 + 16;             // second 16-row subtile
#pragma unroll
    for (int kk = 0; kk < BK; kk += 4) {
      const int kf = kk + 2 * half;           // lanes 0-15: K={kk,kk+1}; 16-31: K={kk+2,kk+3}
      const v2f a0 = *reinterpret_cast<const v2f*>(&As[arow0][kf]);
      const v2f a1 = *reinterpret_cast<const v2f*>(&As[arow1][kf]);
      const v2f b0 = *reinterpret_cast<const v2f*>(&Bs[ 0 + l16][kf]);
      const v2f b1 = *reinterpret_cast<const v2f*>(&Bs[16 + l16][kf]);
      const v2f b2 = *reinterpret_cast<const v2f*>(&Bs[32 + l16][kf]);
      const v2f b3 = *reinterpret_cast<const v2f*>(&Bs[48 + l16][kf]);
      acc[0][0] = __builtin_amdgcn_wmma_f32_16x16x4_f32(false, a0, false, b0, (short)0, acc[0][0], false, false);
      acc[0][1] = __builtin_amdgcn_wmma_f32_16x16x4_f32(false, a0, false, b1, (short)0, acc[0][1], false, false);
      acc[0][2] = __builtin_amdgcn_wmma_f32_16x16x4_f32(false, a0, false, b2, (short)0, acc[0][2], false, false);
      acc[0][3] = __builtin_amdgcn_wmma_f32_16x16x4_f32(false, a0, false, b3, (short)0, acc[0][3], false, false);
      acc[1][0] = __builtin_amdgcn_wmma_f32_16x16x4_f32(false, a1, false, b0, (short)0, acc[1][0], false, false);
      acc[1][1] = __builtin_amdgcn_wmma_f32_16x16x4_f32(false, a1, false, b1, (short)0, acc[1][1], false, false);
      acc[1][2] = __builtin_amdgcn_wmma_f32_16x16x4_f32(false, a1, false, b2, (short)0, acc[1][2], false, false);
      acc[1][3] = __builtin_amdgcn_wmma_f32_16x16x4_f32(false, a1, false, b3, (short)0, acc[1][3], false, false);
    }
    __syncthreads();
  }

  // ---- write back: C VGPR r holds rows {M=r, M=r+8} (half selects), N = lane&15 ----
#pragma unroll
  for (int s = 0; s < 2; ++s) {
    const int mRow0 = mBase + wave * 32 + s * 16 + half * 8;
#pragma unroll
    for (int nt = 0; nt < 4; ++nt) {
      const int col = nBase + nt * 16 + l16;
      const float bv = bias ? bias[col] : 0.0f;
#pragma unroll
      for (int r = 0; r < 8; ++r)
        C[(size_t)(mRow0 + r) * ldc + col] = alpha * acc[s][nt][r] + bv;
    }
  }
}

// ---------------------------------------------------------------------------
// Position init: linspace(-L/2, L/2, L) * scale + bias ; pad -> 10000
// ---------------------------------------------------------------------------
__global__ void pos_init_kernel(const int* __restrict__ lengths,
                                const float* __restrict__ scale,
                                const float* __restrict__ bias,
                                float* __restrict__ pos, int maxLen)
{
  const int b = blockIdx.x;
  const int j = threadIdx.x;
  const int L = lengths[b];
  const float Lf = (float)L;
  const float step = (Lf > 1.0f) ? (Lf / (Lf - 1.0f)) : 0.0f;
  const float p = (-Lf * 0.5f + (float)j * step) * scale[0] + bias[0];
  pos[b * maxLen + j] = (j >= L) ? PAD_POSF : p;
}

// ---------------------------------------------------------------------------
// Per-iteration: da row, accumulate product daP, compute qp_new (row-normalized da @ kp)
// one block per (b, q) row; 256 threads x 2 k's
// ---------------------------------------------------------------------------
__global__ __launch_bounds__(256)
void da_qp_kernel(const float* __restrict__ qp, const float* __restrict__ kp,
                  float* __restrict__ da, float* __restrict__ daP, int firstIter,
                  float* __restrict__ qpNew)
{
  __shared__ float redN[256];
  __shared__ float redD[256];
  const int b = blockIdx.x >> 9;
  const int q = blockIdx.x & 511;
  const int t = threadIdx.x;
  const float qpv = qp[b * LQ + q];
  const bool qmask = qpv >= 1000.0f;
  const size_t rowOff = ((size_t)b * LQ + q) * LK;

  float num = 0.f, den = 0.f;
#pragma unroll
  for (int i = 0; i < 2; ++i) {
    const int k = t + i * 256;
    const float kpv = kp[b * LK + k];
    const float d = qpv - kpv;
    float w = expf(-0.5f * d * d);
    if (qmask || kpv >= 1000.0f) w = 0.f;
    da[rowOff + k] = w;
    daP[rowOff + k] = firstIter ? w : (daP[rowOff + k] * w);
    num += w * kpv;
    den += w;
  }
  redN[t] = num; redD[t] = den;
  __syncthreads();
  for (int s = 128; s > 0; s >>= 1) {
    if (t < s) { redN[t] += redN[t + s]; redD[t] += redD[t + s]; }
    __syncthreads();
  }
  if (t == 0)
    qpNew[b * LQ + q] = qmask ? PAD_POSF : (redN[0] / fmaxf(redD[0], 1e-9f));
}

// kp_new: column-normalized da^T @ qp (coalesced over k)
__global__ __launch_bounds__(256)
void kp_update_kernel(const float* __restrict__ qp, const float* __restrict__ kp,
                      const float* __restrict__ da, float* __restrict__ kpNew)
{
  const int b = blockIdx.y;
  const int k = blockIdx.x * 256 + threadIdx.x;
  const float kpv = kp[b * LK + k];
  const float* daB = da + (size_t)b * LQ * LK;
  float num = 0.f, den = 0.f;
  for (int q = 0; q < LQ; ++q) {
    const float w = daB[(size_t)q * LK + k];
    num += w * qp[b * LQ + q];
    den += w;
  }
  kpNew[b * LK + k] = (kpv >= 1000.0f) ? PAD_POSF : (num / fmaxf(den, 1e-9f));
}

// ---------------------------------------------------------------------------
// Fused finalize: attn = exp(s - rowmax) * daP / sum   (softmax + 3 renorms telescoped)
// one block per (b,h,q) row; 128 threads x 4 k's; in-place over scores
// ---------------------------------------------------------------------------
__global__ __launch_bounds__(128)
void attn_finalize_kernel(float* __restrict__ scores, const float* __restrict__ daP)
{
  __shared__ float red[128];
  const int z = blockIdx.x;            // (b*16+h)*512 + q
  const int q = z & 511;
  const int b = (z >> 9) >> 4;
  const int t = threadIdx.x;
  float* row = scores + (size_t)z * LK;
  const float* dRow = daP + ((size_t)b * LQ + q) * LK;

  float s[4], w[4];
  float m = -3.4e38f;
#pragma unroll
  for (int i = 0; i < 4; ++i) { s[i] = row[t + i * 128]; m = fmaxf(m, s[i]); }
  red[t] = m; __syncthreads();
  for (int st = 64; st > 0; st >>= 1) {
    if (t < st) red[t] = fmaxf(red[t], red[t + st]);
    __syncthreads();
  }
  m = red[0]; __syncthreads();

  float sum = 0.f;
#pragma unroll
  for (int i = 0; i < 4; ++i) { w[i] = expf(s[i] - m) * dRow[t + i * 128]; sum += w[i]; }
  red[t] = sum; __syncthreads();
  for (int st = 64; st > 0; st >>= 1) {
    if (t < st) red[t] += red[t + st];
    __syncthreads();
  }
  const float Z = red[0];
  const float inv = (Z > 1e-30f) ? (1.0f / Z) : 0.0f;
#pragma unroll
  for (int i = 0; i < 4; ++i) row[t + i * 128] = w[i] * inv;
}

// mean over heads -> second output slab
__global__ __launch_bounds__(256)
void attn_mean_kernel(const float* __restrict__ attn, float* __restrict__ outMean)
{
  const size_t idx = (size_t)blockIdx.x * 256 + threadIdx.x;  // over B*LQ*LK
  const size_t k = idx & 511;
  const size_t q = (idx >> 9) & 511;
  const size_t b = idx >> 18;
  float acc = 0.f;
#pragma unroll
  for (int h = 0; h < N_HEADSC; ++h)
    acc += attn[(((b * N_HEADSC + h) * LQ) + q) * LK + k];
  outMean[idx] = acc * (1.0f / N_HEADSC);
}

// ---------------------------------------------------------------------------
extern "C" void kernel_launch(void* const* d_in, const int* in_sizes, int n_in,
                              void* d_out, int out_size, void* d_ws, size_t ws_size,
                              hipStream_t stream) {
  (void)in_sizes; (void)n_in; (void)out_size; (void)ws_size;
  const float* query = (const float*)d_in[0];
  const float* key   = (const float*)d_in[1];
  const float* value = (const float*)d_in[2];
  const int*   qlen  = (const int*)d_in[3];
  const int*   klen  = (const int*)d_in[4];
  const float* Wq = (const float*)d_in[5];  const float* bq = (const float*)d_in[6];
  const float* Wk = (const float*)d_in[7];  const float* bk = (const float*)d_in[8];
  const float* Wv = (const float*)d_in[9];  const float* bv = (const float*)d_in[10];
  const float* Wo = (const float*)d_in[11]; const float* bo = (const float*)d_in[12];
  const float* pscale = (const float*)d_in[13];
  const float* pbias  = (const float*)d_in[14];
  float* out = (float*)d_out;

  const long long QKV  = (long long)BSZ * LQ * D_MODELC;       // 4,194,304
  const long long ATTN = (long long)BSZ * N_HEADSC * LQ * LK;  // 33,554,432
  const long long BLKK = (long long)BSZ * LQ * LK;             // 2,097,152

  float* Qb     = (float*)d_ws;
  float* Kb     = Qb + QKV;
  float* Vb     = Kb + QKV;
  float* scores = Vb + QKV;
  float* ctx    = scores + ATTN;
  float* daP    = ctx + QKV;
  float* da     = daP + BLKK;
  float* qpA    = da + BLKK;
  float* qpB    = qpA + BSZ * LQ;
  float* kpA    = qpB + BSZ * LQ;
  float* kpB    = kpA + BSZ * LK;

  // 1) QKV projections: [4096,1024] @ W^T + b
  dim3 gP(D_MODELC / BN, (BSZ * LQ) / BM, 1);
  wmma_gemm_kernel<<<gP, 128, 0, stream>>>(query, Wq, bq, Qb,
      BSZ * LQ, D_MODELC, D_MODELC, D_MODELC, D_MODELC, D_MODELC, 1, 1.0f, 1,
      0, 0, 0, 0, 0, 0);
  wmma_gemm_kernel<<<gP, 128, 0, stream>>>(key, Wk, bk, Kb,
      BSZ * LQ, D_MODELC, D_MODELC, D_MODELC, D_MODELC, D_MODELC, 1, 1.0f, 1,
      0, 0, 0, 0, 0, 0);
  wmma_gemm_kernel<<<gP, 128, 0, stream>>>(value, Wv, bv, Vb,
      BSZ * LQ, D_MODELC, D_MODELC, D_MODELC, D_MODELC, D_MODELC, 1, 1.0f, 1,
      0, 0, 0, 0, 0, 0);

  // 2) scores = Qh @ Kh^T / 8, batched over (b,h)
  dim3 gS(LK / BN, LQ / BM, BSZ * N_HEADSC);
  wmma_gemm_kernel<<<gS, 128, 0, stream>>>(Qb, Kb, nullptr, scores,
      LQ, LK, D_KC, D_MODELC, D_MODELC, LK, 1, 0.125f, N_HEADSC,
      (long long)LQ * D_MODELC, D_KC,
      (long long)LK * D_MODELC, D_KC,
      (long long)N_HEADSC * LQ * LK, (long long)LQ * LK);

  // 3) positions
  pos_init_kernel<<<BSZ, LQ, 0, stream>>>(qlen, pscale, pbias, qpA, LQ);
  pos_init_kernel<<<BSZ, LK, 0, stream>>>(klen, pscale, pbias, kpA, LK);

  // 4) three position iterations, accumulating daP = da1*da2*da3
  float* qpCur = qpA; float* qpNxt = qpB;
  float* kpCur = kpA; float* kpNxt = kpB;
  for (int it = 0; it < 3; ++it) {
    da_qp_kernel<<<BSZ * LQ, 256, 0, stream>>>(qpCur, kpCur, da, daP, it == 0, qpNxt);
    kp_update_kernel<<<dim3(LK / 256, BSZ), 256, 0, stream>>>(qpCur, kpCur, da, kpNxt);
    float* t0 = qpCur; qpCur = qpNxt; qpNxt = t0;
    float* t1 = kpCur; kpCur = kpNxt; kpNxt = t1;
  }

  // 5) finalize attention in place over scores
  attn_finalize_kernel<<<BSZ * N_HEADSC * LQ, 128, 0, stream>>>(scores, daP);

  // 6) ctx = attn @ Vh, batched over (b,h); B non-transposed
  dim3 gC(D_KC / BN, LQ / BM, BSZ * N_HEADSC);
  wmma_gemm_kernel<<<gC, 128, 0, stream>>>(scores, Vb, nullptr, ctx,
      LQ, D_KC, LK, LK, D_MODELC, D_MODELC, 0, 1.0f, N_HEADSC,
      (long long)N_HEADSC * LQ * LK, (long long)LQ * LK,
      (long long)LK * D_MODELC, D_KC,
      (long long)LQ * D_MODELC, D_KC);

  // 7) out = ctx @ Wo^T + bo  -> first output slab
  wmma_gemm_kernel<<<gP, 128, 0, stream>>>(ctx, Wo, bo, out,
      BSZ * LQ, D_MODELC, D_MODELC, D_MODELC, D_MODELC, D_MODELC, 1, 1.0f, 1,
      0, 0, 0, 0, 0, 0);

  // 8) mean over heads -> second output slab
  attn_mean_kernel<<<(BSZ * LQ * LK) / 256, 256, 0, stream>>>(scores, out + QKV);
}